// BatchMemoryWrapLayer_12610023981072
// MI455X (gfx1250) — compile-verified
//
#include <hip/hip_runtime.h>
#include <hip/hip_bf16.h>
#include <math.h>

typedef float v2f __attribute__((ext_vector_type(2)));
typedef float v4f __attribute__((ext_vector_type(4)));
typedef float v8f __attribute__((ext_vector_type(8)));

#define B_SZ    64
#define N_SZ    4096
#define D_SZ    1024
#define DIN_SZ  2048
#define DHID_SZ 4096
#define OUT_SZ  1000
#define EPSF    1e-6f
#define NCHUNK  8                 // split of N for the weighted-sum pass
#define CHN     (N_SZ / NCHUNK)   // 512

// ---------------------------------------------------------------------------
// Kernel 1: per-row L2-normalize encoder_output -> xn, and copy enc into the
// first half of h_in = concat(enc, mv).   grid = B, block = 256
// ---------------------------------------------------------------------------
__global__ void k_norm_enc(const float* __restrict__ enc,
                           float* __restrict__ xn,
                           float* __restrict__ h_in) {
  const int b = blockIdx.x;
  const int t = threadIdx.x;
  const float* e = enc + (size_t)b * D_SZ;
  __shared__ float red[256];

  float ss = 0.f;
  for (int i = t; i < D_SZ; i += 256) { float v = e[i]; ss += v * v; }
  red[t] = ss; __syncthreads();
  for (int s = 128; s; s >>= 1) { if (t < s) red[t] += red[t + s]; __syncthreads(); }
  const float inv = 1.0f / fmaxf(sqrtf(red[0]), EPSF);
  __syncthreads();

  for (int i = t; i < D_SZ; i += 256) {
    float v = e[i];
    xn[(size_t)b * D_SZ + i] = v * inv;
    h_in[(size_t)b * DIN_SZ + i] = v;  // first half of concat
  }
}

// ---------------------------------------------------------------------------
// Kernel 2: z[b,n] = cos(xn[b], mem[b,n]) - 1   (== -dist)
// One wave32 per memory row; NT-hinted float4 streaming of 1 GiB (dominant
// cost; no reuse, keep it out of L2). xn row is reused -> regular loads.
// grid = B*N/8, block = 256 (8 waves)
// ---------------------------------------------------------------------------
__global__ void k_scores(const float* __restrict__ mem,
                         const float* __restrict__ xn,
                         float* __restrict__ z) {
  const int tid  = threadIdx.x;
  const int wave = tid >> 5;
  const int lane = tid & 31;
  const int row  = blockIdx.x * 8 + wave;        // [0, B*N)
  const int b = row >> 12;                        // / N_SZ
  const v4f* mrow = (const v4f*)(mem + (size_t)row * D_SZ);
  const v4f* xrow = (const v4f*)(xn + (size_t)b * D_SZ);

  float dot = 0.f, ss = 0.f;
#pragma unroll
  for (int i = 0; i < 8; ++i) {
    const int idx = i * 32 + lane;               // 256 float4 per row, coalesced
    v4f m = __builtin_nontemporal_load(&mrow[idx]);
    v4f x = xrow[idx];
    dot += m[0] * x[0] + m[1] * x[1] + m[2] * x[2] + m[3] * x[3];
    ss  += m[0] * m[0] + m[1] * m[1] + m[2] * m[2] + m[3] * m[3];
  }
#pragma unroll
  for (int off = 16; off; off >>= 1) {
    dot += __shfl_xor(dot, off, 32);
    ss  += __shfl_xor(ss,  off, 32);
  }
  if (lane == 0) {
    z[row] = dot / fmaxf(sqrtf(ss), EPSF) - 1.0f;
  }
}

// ---------------------------------------------------------------------------
// Kernel 3: sparsemax per batch row (no sort).
// tau solves sum(relu(z - tau)) = 1 (monotone, piecewise linear). Bisect in
// [max(z)-1, max(z)], then recover the exact reference tau from the support.
// grid = B, block = 256
// ---------------------------------------------------------------------------
__global__ void k_sparsemax(const float* __restrict__ z_all,
                            float* __restrict__ w_all) {
  const int b = blockIdx.x;
  const int t = threadIdx.x;
  const float* z = z_all + (size_t)b * N_SZ;
  float* w = w_all + (size_t)b * N_SZ;

  __shared__ float zs[N_SZ];     // 16 KB
  __shared__ float red[256];

  for (int i = t; i < N_SZ; i += 256) zs[i] = z[i];
  __syncthreads();

  // row max
  float m = -INFINITY;
  for (int i = t; i < N_SZ; i += 256) m = fmaxf(m, zs[i]);
  red[t] = m; __syncthreads();
  for (int s = 128; s; s >>= 1) { if (t < s) red[t] = fmaxf(red[t], red[t + s]); __syncthreads(); }
  const float zmax = red[0];
  __syncthreads();

  float lo = zmax - 1.0f;   // f(lo) >= 0
  float hi = zmax;          // f(hi) = -1 < 0
  for (int iter = 0; iter < 36; ++iter) {
    const float mid = 0.5f * (lo + hi);
    float sum = 0.f;
    for (int i = t; i < N_SZ; i += 256) sum += fmaxf(zs[i] - mid, 0.f);
    red[t] = sum; __syncthreads();
    for (int s = 128; s; s >>= 1) { if (t < s) red[t] += red[t + s]; __syncthreads(); }
    const float f = red[0] - 1.0f;
    __syncthreads();
    if (f >= 0.f) lo = mid; else hi = mid;     // block-uniform branch
  }

  // exact tau from support {z > lo}
  float cnt = 0.f, ssum = 0.f;
  for (int i = t; i < N_SZ; i += 256) {
    float v = zs[i];
    if (v > lo) { cnt += 1.0f; ssum += v; }
  }
  red[t] = cnt; __syncthreads();
  for (int s = 128; s; s >>= 1) { if (t < s) red[t] += red[t + s]; __syncthreads(); }
  const float k = red[0]; __syncthreads();
  red[t] = ssum; __syncthreads();
  for (int s = 128; s; s >>= 1) { if (t < s) red[t] += red[t + s]; __syncthreads(); }
  const float sk = red[0]; __syncthreads();

  const float tau = (sk - 1.0f) / fmaxf(k, 1.0f);
  for (int i = t; i < N_SZ; i += 256) w[i] = fmaxf(zs[i] - tau, 0.f);
}

// ---------------------------------------------------------------------------
// Kernel 4a: per-chunk partial weighted sum over a 512-row slice of mem.
// Skips zero weights (block-uniform branch off an LDS broadcast), so typical
// cost is a few rows; worst case 512 blocks stream the full 1 GiB.
// grid = B * NCHUNK, block = 256 (thread t owns float4 at d = 4t)
// partial layout: [B][NCHUNK][D]
// ---------------------------------------------------------------------------
__global__ void k_wsum_partial(const float* __restrict__ mem,
                               const float* __restrict__ w_all,
                               float* __restrict__ partial) {
  const int b  = blockIdx.x / NCHUNK;
  const int ch = blockIdx.x % NCHUNK;
  const int t  = threadIdx.x;
  const float* w = w_all + (size_t)b * N_SZ + (size_t)ch * CHN;
  const v4f* memb = (const v4f*)(mem + ((size_t)b * N_SZ + (size_t)ch * CHN) * D_SZ);
  __shared__ float ws[256];

  v4f acc = {0.f, 0.f, 0.f, 0.f};
  for (int n0 = 0; n0 < CHN; n0 += 256) {
    ws[t] = w[n0 + t];
    __syncthreads();
    for (int j = 0; j < 256; ++j) {
      const float wj = ws[j];                  // LDS broadcast, uniform branch
      if (wj != 0.f) {
        v4f m = __builtin_nontemporal_load(&memb[(size_t)(n0 + j) * (D_SZ / 4) + t]);
        acc += wj * m;
      }
    }
    __syncthreads();
  }
  v4f* dst = (v4f*)(partial + ((size_t)b * NCHUNK + ch) * D_SZ);
  dst[t] = acc;
}

// ---------------------------------------------------------------------------
// Kernel 4b: reduce the NCHUNK partials in fixed chunk order (deterministic)
// into the second half of h_in.   grid = B, block = 256
// ---------------------------------------------------------------------------
__global__ void k_wsum_reduce(const float* __restrict__ partial,
                              float* __restrict__ h_in) {
  const int b = blockIdx.x;
  const int t = threadIdx.x;
  v4f acc = {0.f, 0.f, 0.f, 0.f};
#pragma unroll
  for (int ch = 0; ch < NCHUNK; ++ch) {
    const v4f* src = (const v4f*)(partial + ((size_t)b * NCHUNK + ch) * D_SZ);
    acc += src[t];
  }
  v4f* dst = (v4f*)(h_in + (size_t)b * DIN_SZ + D_SZ);
  dst[t] = acc;
}

// ---------------------------------------------------------------------------
// Kernel 5: h = relu(h_in @ W1^T + b1)  -- [64,2048]x[2048,4096]
// V_WMMA_F32_16X16X4_F32, one 16x16 tile per wave, K stepped by 4.
// A frag: lane(0-15)=row M, VGPRs = K{0,1}; lanes 16-31 = K{2,3}  -> float2
// B frag: lane(0-15)=col N, VGPRs = K{0,1}; lanes 16-31 = K{2,3}  -> float2
//         B[k][n] = W1[n][k] so both loads are contiguous along K.
// grid = (4*256)/8 = 128 blocks, block = 256 (8 waves)
// ---------------------------------------------------------------------------
__global__ void k_gemm1_relu(const float* __restrict__ A,   // h_in [64,2048]
                             const float* __restrict__ W1,  // [4096,2048]
                             const float* __restrict__ b1,
                             float* __restrict__ H) {       // [64,4096]
  const int tid  = threadIdx.x;
  const int wave = tid >> 5;
  const int lane = tid & 31;
  const int gtile = blockIdx.x * 8 + wave;
  const int TN = DHID_SZ / 16;                 // 256
  const int tm = gtile / TN;
  const int tn = gtile % TN;
  const int half = lane >> 4;
  const int l15  = lane & 15;

  const float* arow = A  + (size_t)(tm * 16 + l15) * DIN_SZ;
  const float* brow = W1 + (size_t)(tn * 16 + l15) * DIN_SZ;

  v8f c = {0.f, 0.f, 0.f, 0.f, 0.f, 0.f, 0.f, 0.f};
#pragma unroll 8
  for (int k0 = 0; k0 < DIN_SZ; k0 += 4) {
    v2f a = *(const v2f*)(arow + k0 + 2 * half);
    v2f bb = *(const v2f*)(brow + k0 + 2 * half);
    c = __builtin_amdgcn_wmma_f32_16x16x4_f32(false, a, false, bb,
                                              (short)0, c, false, false);
  }

  const int col = tn * 16 + l15;
  const float bias = b1[col];
#pragma unroll
  for (int v = 0; v < 8; ++v) {
    const int rowm = tm * 16 + v + 8 * half;   // C/D: M = vgpr + 8*half
    H[(size_t)rowm * DHID_SZ + col] = fmaxf(c[v] + bias, 0.f);
  }
}

// ---------------------------------------------------------------------------
// Kernel 6: out = H @ W2^T + b2  -- [64,4096]x[4096,1000], N padded to 1008.
// grid = 32 blocks * 8 waves = 256 waves, 252 active tiles (4 x 63)
// ---------------------------------------------------------------------------
__global__ void k_gemm2(const float* __restrict__ H,    // [64,4096]
                        const float* __restrict__ W2,   // [1000,4096]
                        const float* __restrict__ b2,
                        float* __restrict__ out) {      // [64,1000]
  const int tid  = threadIdx.x;
  const int wave = tid >> 5;
  const int lane = tid & 31;
  const int gtile = blockIdx.x * 8 + wave;
  const int TN = (OUT_SZ + 15) / 16;           // 63
  if (gtile >= 4 * TN) return;                 // wave-uniform exit
  const int tm = gtile / TN;
  const int tn = gtile % TN;
  const int half = lane >> 4;
  const int l15  = lane & 15;

  const int ncol = tn * 16 + l15;              // this lane's output column
  const bool nvalid = (ncol < OUT_SZ);
  const float* arow = H + (size_t)(tm * 16 + l15) * DHID_SZ;
  const float* brow = W2 + (size_t)(nvalid ? ncol : 0) * DHID_SZ;

  v8f c = {0.f, 0.f, 0.f, 0.f, 0.f, 0.f, 0.f, 0.f};
#pragma unroll 8
  for (int k0 = 0; k0 < DHID_SZ; k0 += 4) {
    v2f a = *(const v2f*)(arow + k0 + 2 * half);
    v2f bb;
    if (nvalid) bb = *(const v2f*)(brow + k0 + 2 * half);
    else        { bb[0] = 0.f; bb[1] = 0.f; }
    c = __builtin_amdgcn_wmma_f32_16x16x4_f32(false, a, false, bb,
                                              (short)0, c, false, false);
  }

  if (nvalid) {
    const float bias = b2[ncol];
#pragma unroll
    for (int v = 0; v < 8; ++v) {
      const int rowm = tm * 16 + v + 8 * half;
      out[(size_t)rowm * OUT_SZ + ncol] = c[v] + bias;
    }
  }
}

// ---------------------------------------------------------------------------
// Host-side launcher
// ---------------------------------------------------------------------------
extern "C" void kernel_launch(void* const* d_in, const int* in_sizes, int n_in,
                              void* d_out, int out_size, void* d_ws, size_t ws_size,
                              hipStream_t stream) {
  const float* enc = (const float*)d_in[0];   // [64,1024]
  const float* mem = (const float*)d_in[1];   // [64,4096,1024]
  const float* W1  = (const float*)d_in[2];   // [4096,2048]
  const float* b1  = (const float*)d_in[3];   // [4096]
  const float* W2  = (const float*)d_in[4];   // [1000,4096]
  const float* b2  = (const float*)d_in[5];   // [1000]
  float* out = (float*)d_out;                 // [64,1000]

  float* ws = (float*)d_ws;
  float* xn      = ws;                                     // 64*1024      = 65536
  float* z       = xn + (size_t)B_SZ * D_SZ;               // 64*4096      = 262144
  float* wsp     = z + (size_t)B_SZ * N_SZ;                // 64*4096      = 262144
  float* h_in    = wsp + (size_t)B_SZ * N_SZ;              // 64*2048      = 131072
  float* h       = h_in + (size_t)B_SZ * DIN_SZ;           // 64*4096      = 262144
  float* partial = h + (size_t)B_SZ * DHID_SZ;             // 64*8*1024    = 524288
  // total: ~1.5M floats = 6 MB of workspace

  k_norm_enc<<<B_SZ, 256, 0, stream>>>(enc, xn, h_in);
  k_scores<<<(B_SZ * N_SZ) / 8, 256, 0, stream>>>(mem, xn, z);
  k_sparsemax<<<B_SZ, 256, 0, stream>>>(z, wsp);
  k_wsum_partial<<<B_SZ * NCHUNK, 256, 0, stream>>>(mem, wsp, partial);
  k_wsum_reduce<<<B_SZ, 256, 0, stream>>>(partial, h_in);
  k_gemm1_relu<<<(4 * (DHID_SZ / 16)) / 8, 256, 0, stream>>>(h_in, W1, b1, h);
  k_gemm2<<<32, 256, 0, stream>>>(h, W2, b2, out);
}